// AnomalyAttention_47802986004962
// MI455X (gfx1250) — compile-verified
//
#include <hip/hip_runtime.h>
#include <hip/hip_bf16.h>

typedef __attribute__((ext_vector_type(2))) float v2f;
typedef __attribute__((ext_vector_type(4))) float v4f;
typedef __attribute__((ext_vector_type(8))) float v8f;

// Problem constants from the reference
#define BB 8
#define LL 1024
#define SS 1024
#define HH 8
#define EE 64
#define DD 64
#define SCALE 0.125f        // 1/sqrt(64)
#define NEGBIG (-3.0e38f)

// XOR-swizzled LDS row buffer: 16 rows x 1024 cols = exactly 64 KB.
// Swizzle flips bits [5:2] only -> quads stay contiguous (b64/b128 friendly),
// and 16 lanes reading fixed k across m hit 16 distinct banks.
#define SIDX(m, k) (((m) << 10) + ((k) ^ ((m) << 2)))

__global__ __launch_bounds__(128) void
anomaly_attn_kernel(const float* __restrict__ Q,
                    const float* __restrict__ K,
                    const float* __restrict__ Vv,
                    float* __restrict__ outV,
                    float* __restrict__ outSeries) {
    __shared__ float srow[16 * 1024];   // scores -> probs for this 16-row block

    const int it = blockIdx.x;          // row-tile index (0..63)
    const int h  = blockIdx.y;
    const int b  = blockIdx.z;
    const int i0 = it * 16;

    const int t    = threadIdx.x;
    const int wave = t >> 5;            // 0..3
    const int lane = t & 31;
    const int half = lane >> 4;         // selects K-pair within WMMA frag
    const int mn   = lane & 15;         // M index for A, N index for B/C/D

    // Q/K element: [((b*L + row)*H + h)*E + e]; row stride = H*E = 512 floats
    const size_t qkbase = ((size_t)b * LL * HH + h) * EE;

    // ---------- Phase 1: scores = scale * Q K^T, causal-masked, into LDS ----
    for (int jt = wave; jt <= it; jt += 4) {
        const int j0 = jt * 16;
        // prefetch this wave's next K tile (16 rows covered by lanes 0..15)
        if (jt + 4 <= it)
            __builtin_prefetch(K + qkbase + (size_t)(j0 + 64 + mn) * 512, 0, 0);
        v8f c = {};
#pragma unroll
        for (int ec = 0; ec < 16; ++ec) {          // E=64 in K-chunks of 4
            const int kb = ec * 4 + half * 2;
            const v2f a  = *(const v2f*)(Q + qkbase + (size_t)(i0 + mn) * 512 + kb);
            const v2f bb = *(const v2f*)(K + qkbase + (size_t)(j0 + mn) * 512 + kb);
            c = __builtin_amdgcn_wmma_f32_16x16x4_f32(
                    false, a, false, bb, (short)0, c, false, false);
        }
#pragma unroll
        for (int v = 0; v < 8; ++v) {
            const int m = v + 8 * half;
            const int i = i0 + m;
            const int j = j0 + mn;
            float x = c[v] * SCALE;
            if (j > i) x = NEGBIG;                 // TriangularCausalMask
            srow[SIDX(m, j)] = x;
        }
    }
    __syncthreads();

    // ---------- Phase 2: softmax; wave w owns rows 4w..4w+3 (whole wave/row)
    const int jlimit = (it + 1) * 16;              // cols >= jlimit fully masked
    for (int rr = 0; rr < 4; ++rr) {
        const int row = (wave << 2) + rr;

        float mx = NEGBIG;
        for (int j = lane; j < jlimit; j += 32)
            mx = fmaxf(mx, srow[SIDX(row, j)]);
        mx = fmaxf(mx, __shfl_xor(mx, 1, 32));
        mx = fmaxf(mx, __shfl_xor(mx, 2, 32));
        mx = fmaxf(mx, __shfl_xor(mx, 4, 32));
        mx = fmaxf(mx, __shfl_xor(mx, 8, 32));
        mx = fmaxf(mx, __shfl_xor(mx, 16, 32));

        float sum = 0.0f;
        for (int j = lane; j < jlimit; j += 32) {
            const float e = __expf(srow[SIDX(row, j)] - mx);
            srow[SIDX(row, j)] = e;
            sum += e;
        }
        sum += __shfl_xor(sum, 1, 32);
        sum += __shfl_xor(sum, 2, 32);
        sum += __shfl_xor(sum, 4, 32);
        sum += __shfl_xor(sum, 8, 32);
        sum += __shfl_xor(sum, 16, 32);
        const float rinv = 1.0f / sum;

        // normalize + stream out: 512B contiguous per wave store (b128/lane)
        const size_t sb = (((size_t)b * HH + h) * LL + (i0 + row)) * SS;
        for (int j4 = lane * 4; j4 < SS; j4 += 128) {
            v4f p = {0.0f, 0.0f, 0.0f, 0.0f};
            if (j4 < jlimit) {                     // jlimit % 16 == 0: whole quad
                p.x = srow[SIDX(row, j4 + 0)] * rinv;
                p.y = srow[SIDX(row, j4 + 1)] * rinv;
                p.z = srow[SIDX(row, j4 + 2)] * rinv;
                p.w = srow[SIDX(row, j4 + 3)] * rinv;
                srow[SIDX(row, j4 + 0)] = p.x;
                srow[SIDX(row, j4 + 1)] = p.y;
                srow[SIDX(row, j4 + 2)] = p.z;
                srow[SIDX(row, j4 + 3)] = p.w;
            }
            __builtin_nontemporal_store(p, (v4f*)(outSeries + sb + j4));
        }
    }
    __syncthreads();

    // ---------- Phase 3: V = P * values (wave w owns D-tile w) --------------
    const int d0 = wave * 16;
    const size_t vbase = (size_t)b * SS * HH * DD + (size_t)h * DD; // + s*512 + d
    v8f acc = {};
    const int kcmax = (it + 1) * 4;                // causal-truncated K loop
    for (int kc = 0; kc < kcmax; ++kc) {
        const int kb = kc * 4 + half * 2;
        v2f a, bb;
        a.x  = srow[SIDX(mn, kb)];
        a.y  = srow[SIDX(mn, kb + 1)];             // contiguous -> ds_load_b64
        bb.x = Vv[vbase + (size_t)(kb)     * 512 + d0 + mn];
        bb.y = Vv[vbase + (size_t)(kb + 1) * 512 + d0 + mn];
        acc = __builtin_amdgcn_wmma_f32_16x16x4_f32(
                  false, a, false, bb, (short)0, acc, false, false);
    }
#pragma unroll
    for (int v = 0; v < 8; ++v) {
        const int m = v + 8 * half;
        const int i = i0 + m;
        __builtin_nontemporal_store(
            acc[v], outV + (((size_t)b * LL + i) * HH + h) * DD + d0 + mn);
    }
}

// prior[b,i,s] = sum_h exp(-|i-s|/sigma[b,h,s]), row-normalized (1/H cancels)
__global__ __launch_bounds__(256) void
anomaly_prior_kernel(const float* __restrict__ sigma,
                     float* __restrict__ outPrior) {
    const int i = blockIdx.x;
    const int b = blockIdx.y;
    const int t = threadIdx.x;

    v4f vals;
    float psum = 0.0f;
#pragma unroll
    for (int u = 0; u < 4; ++u) {
        const int s = t * 4 + u;                   // 4 contiguous cols per thread
        const float d = fabsf((float)(i - s));
        float acc = 0.0f;
#pragma unroll
        for (int hh = 0; hh < HH; ++hh) {
            const float sg = sigma[((size_t)b * HH + hh) * LL + s];
            acc += __expf(-d / sg);
        }
        vals[u] = acc;
        psum += acc;
    }

    __shared__ float red[256];
    red[t] = psum;
    __syncthreads();
    for (int off = 128; off > 0; off >>= 1) {
        if (t < off) red[t] += red[t + off];
        __syncthreads();
    }
    const float inv = 1.0f / red[0];

    const v4f p = {vals.x * inv, vals.y * inv, vals.z * inv, vals.w * inv};
    const size_t pbase = ((size_t)b * LL + i) * SS;
    __builtin_nontemporal_store(p, (v4f*)(outPrior + pbase + t * 4));
}

extern "C" void kernel_launch(void* const* d_in, const int* in_sizes, int n_in,
                              void* d_out, int out_size, void* d_ws, size_t ws_size,
                              hipStream_t stream) {
    const float* Q     = (const float*)d_in[0];
    const float* K     = (const float*)d_in[1];
    const float* V     = (const float*)d_in[2];
    const float* sigma = (const float*)d_in[3];

    float* out       = (float*)d_out;
    float* outV      = out;                                   // B*L*H*D
    float* outSeries = out + (size_t)BB * LL * HH * DD;       // B*H*L*S
    float* outPrior  = outSeries + (size_t)BB * HH * LL * SS; // B*L*S

    dim3 gridA(LL / 16, HH, BB);   // (64, 8, 8)
    anomaly_attn_kernel<<<gridA, 128, 0, stream>>>(Q, K, V, outV, outSeries);

    dim3 gridB(LL, BB);            // (1024, 8)
    anomaly_prior_kernel<<<gridB, 256, 0, stream>>>(sigma, outPrior);
}